// Model_flow_13606456394323
// MI455X (gfx1250) — compile-verified
//
#include <hip/hip_runtime.h>
#include <stdint.h>

// Problem constants (from reference setup_inputs)
#define BB 8
#define HH 384
#define WW 1280
#define CC 8

// Tile configuration: 32x16 source pixels per 256-thread workgroup.
#define TI 16
#define TJ 32
#define HALO 8
#define IDIM (TI + 2 * HALO + 1)          // 33 destination rows covered
#define JDIM (TJ + 2 * HALO + 1)          // 49 destination cols covered
#define LDS_FLOATS (IDIM * JDIM * CC)     // 12936 floats = 51744 B
#define FLO_TILE (TI * TJ * 2)            // 1024 floats = 4096 B
#define THREADS 256

typedef unsigned int v4u __attribute__((ext_vector_type(4)));
typedef int v8i __attribute__((ext_vector_type(8)));
typedef int v4i __attribute__((ext_vector_type(4)));

__global__ __launch_bounds__(THREADS) void splat_kernel(
    const float* __restrict__ U, const float* __restrict__ flo,
    float* __restrict__ out) {
  __shared__ float acc[LDS_FLOATS];
  __shared__ float flo_lds[FLO_TILE];     // [TI][TJ*2] staged flow tile

  const int tid = threadIdx.x;
  const int tj0 = blockIdx.x * TJ;
  const int ti0 = blockIdx.y * TI;
  const int b = blockIdx.z;
  const int jbase = tj0 - HALO;
  const int ibase = ti0 - HALO;

  // ---- Stage the flo tile into LDS via the Tensor Data Mover (wave 0),
  //      overlapped with accumulator zeroing by all waves. ----
#if __has_builtin(__builtin_amdgcn_tensor_load_to_lds)
  if (tid < 32) {  // one wave issues the DMA; EXEC is ignored by TDM
    unsigned lds_addr = (unsigned)(uintptr_t)flo_lds;  // low 32 bits = LDS byte addr
    uint64_t ga =
        (uint64_t)(uintptr_t)(flo + ((size_t)(b * HH + ti0) * WW + tj0) * 2);
    const unsigned td0 = WW * 2;      // tensor_dim0 (f32 elements)  = 2560
    const unsigned td1 = HH;          // tensor_dim1                 = 384
    const unsigned tile0 = TJ * 2;    // tile_dim0 = 64 contiguous f32 per row
    const unsigned tile1 = TI;        // tile_dim1 = 16 rows
    const unsigned str0 = WW * 2;     // row stride in elements      = 2560
    // D# group 0: count=1 | lds_addr | global_addr[56:0] | type=2 ("image")
    v4u g0 = {1u, lds_addr, (unsigned)(ga & 0xffffffffu),
              (unsigned)((ga >> 32) & 0x01ffffffu) | (2u << 30)};
    // D# group 1 per ISA bit layout (data_size=2 -> 4B elements)
    v8i g1 = {(int)(2u << 16),
              (int)((td0 & 0xffffu) << 16),
              (int)(((td0 >> 16) & 0xffffu) | ((td1 & 0xffffu) << 16)),
              (int)(((td1 >> 16) & 0xffffu) | (tile0 << 16)),
              (int)tile1,            // tile_dim1; tile_dim2 = 0 (2D tile)
              (int)str0,             // tensor_dim0_stride[31:0]
              0,                     // stride0 hi16 = 0; stride1 lo16 = 0
              0};
    v4i gz4 = {0, 0, 0, 0};          // groups 2/3 unused for a 2D tile
#if defined(__clang_major__) && __clang_major__ >= 23
    v8i gz8 = {0, 0, 0, 0, 0, 0, 0, 0};
    __builtin_amdgcn_tensor_load_to_lds(g0, g1, gz4, gz4, gz8, 0);
#else
    __builtin_amdgcn_tensor_load_to_lds(g0, g1, gz4, gz4, 0);
#endif
    __builtin_amdgcn_s_wait_tensorcnt(0);
  }
#else
  for (int t = tid; t < FLO_TILE; t += THREADS) {
    int r = t >> 6, w = t & 63;
    flo_lds[t] = flo[((size_t)(b * HH + ti0 + r) * WW + tj0) * 2 + w];
  }
#endif

  // Zero the LDS accumulator tile (runs while the TDM transfer is in flight).
  for (int t = tid; t < LDS_FLOATS; t += THREADS) acc[t] = 0.0f;
  __syncthreads();

  const float stepW = 2.0f / (WW - 1.0f);
  const float stepH = 2.0f / (HH - 1.0f);
  const float halfW = (WW - 1.0f) * 0.5f;
  const float halfH = (HH - 1.0f) * 0.5f;

  const float2* flds2 = (const float2*)flo_lds;

  // 512 pixels * 8 channels = 4096 items; channel-innermost so 8 consecutive
  // lanes hit 8 consecutive floats (coalesced U loads, conflict-free LDS
  // atomics, coalesced global atomics).
  for (int k = 0; k < 16; ++k) {
    int item = tid + k * THREADS;
    int c = item & (CC - 1);
    int p = item >> 3;          // 0..511 local pixel
    int lj = p & (TJ - 1);
    int li = p >> 5;
    int j = tj0 + lj;
    int i = ti0 + li;
    int pix = (b * HH + i) * WW + j;

    float2 f = flds2[li * TJ + lj];   // ds_load_b64, broadcast across 8 lanes
    float fx = f.x, fy = f.y;

    // Replicate the reference's normalized-grid round trip in fp32 so
    // floor() decisions match (algebraically x = j + fx).
    float xs = -1.0f + (float)j * stepW;
    float ys = -1.0f + (float)i * stepH;
    float xt = xs + fx / halfW;
    float yt = ys + fy / halfH;
    float x = (xt + 1.0f) * halfW;
    float y = (yt + 1.0f) * halfH;

    int ix0 = (int)floorf(x);
    int iy0 = (int)floorf(y);
    int ix1 = ix0 + 1;
    int iy1 = iy0 + 1;
    float x0f = (float)ix0, y0f = (float)iy0;
    float wx0 = (x0f + 1.0f) - x;
    float wx1 = x - x0f;
    float wy0 = (y0f + 1.0f) - y;
    float wy1 = y - y0f;
    bool vx0 = (unsigned)ix0 < (unsigned)WW;
    bool vx1 = (unsigned)ix1 < (unsigned)WW;
    bool vy0 = (unsigned)iy0 < (unsigned)HH;
    bool vy1 = (unsigned)iy1 < (unsigned)HH;

    // U is streamed exactly once -> non-temporal, keep L2 for the atomic set.
    float u = __builtin_nontemporal_load(U + (size_t)pix * CC + c);

    auto tap = [&](int ix, int iy, float w) {
      if (w != 0.0f) {            // w!=0 implies tap is globally in-bounds
        float val = u * w;
        int di = iy - ibase;
        int dj = ix - jbase;
        if ((unsigned)di < (unsigned)IDIM && (unsigned)dj < (unsigned)JDIM) {
          atomicAdd(&acc[(di * JDIM + dj) * CC + c], val);  // ds_add_f32
        } else {
          unsafeAtomicAdd(&out[((b * HH + iy) * WW + ix) * CC + c], val);
        }
      }
    };
    tap(ix0, iy0, (vx0 && vy0) ? wx0 * wy0 : 0.0f);
    tap(ix0, iy1, (vx0 && vy1) ? wx0 * wy1 : 0.0f);
    tap(ix1, iy0, (vx1 && vy0) ? wx1 * wy0 : 0.0f);
    tap(ix1, iy1, (vx1 && vy1) ? wx1 * wy1 : 0.0f);
  }
  __syncthreads();

  // Flush nonzero LDS cells to global with coalesced fp32 atomics.
  // Halo regions overlap between neighboring tiles, so adds must be atomic.
  for (int t = tid; t < LDS_FLOATS; t += THREADS) {
    float v = acc[t];
    if (v != 0.0f) {
      int c = t & (CC - 1);
      int lin = t >> 3;
      int dj = lin % JDIM;
      int di = lin / JDIM;
      int jd = jbase + dj;
      int id = ibase + di;
      if ((unsigned)jd < (unsigned)WW && (unsigned)id < (unsigned)HH) {
        unsafeAtomicAdd(&out[((b * HH + id) * WW + jd) * CC + c], v);
      }
    }
  }
}

__global__ void zero_out_kernel(float4* __restrict__ out, int n4) {
  int idx = blockIdx.x * blockDim.x + threadIdx.x;
  if (idx < n4) out[idx] = make_float4(0.f, 0.f, 0.f, 0.f);
}

extern "C" void kernel_launch(void* const* d_in, const int* in_sizes, int n_in,
                              void* d_out, int out_size, void* d_ws,
                              size_t ws_size, hipStream_t stream) {
  const float* U = (const float*)d_in[0];     // [B,H,W,C] fp32
  const float* flo = (const float*)d_in[1];   // [B,H,W,2] fp32
  float* out = (float*)d_out;                 // [B,H,W,C] fp32

  // 1) Zero the output (harness poisons it; splat accumulates into it).
  int n4 = out_size / 4;                      // 31457280 / 4
  zero_out_kernel<<<(n4 + 255) / 256, 256, 0, stream>>>((float4*)out, n4);

  // 2) Tiled forward splat with TDM-staged flow + LDS aggregation.
  dim3 grid(WW / TJ, HH / TI, BB);            // 40 x 24 x 8 workgroups
  splat_kernel<<<grid, THREADS, 0, stream>>>(U, flo, out);
}